// SpatialPool_75230647157530
// MI455X (gfx1250) — compile-verified
//
#include <hip/hip_runtime.h>

// SpatialPool: out[b][h*38+w][k*512+c] = fm[b][c][clamp(h+kh-1)][clamp(w+kw-1)]
// Pure data movement (~473 MB HBM traffic, 0 FLOPs) -> use the CDNA5 async
// global<->LDS DMA path to do the NCHW->NHWC transpose through LDS.

namespace {
constexpr int BB  = 16;
constexpr int CC  = 512;
constexpr int HW  = 38;           // H == W
constexpr int PP  = HW * HW;      // 1444
constexpr int KK  = 9;
constexpr int OC  = KK * CC;      // 4608
constexpr int CT  = 128;          // channel tile per block
constexpr int CPAD = 132;         // padded channel stride in LDS (floats), %4==0
constexpr int ROWS = 3;           // h-1, h, h+1 (clamped)
constexpr int NSP  = ROWS * HW;   // 114 spatial slots
constexpr int NELEM = NSP * CT;   // 14592 dword async loads (= 57 * 256)
constexpr int NOUT  = HW * KK * (CT / 4); // 10944 b128 async stores
constexpr int NTHREADS = 256;     // 8 waves (wave32)
}

__global__ __launch_bounds__(NTHREADS)
void SpatialPool_gather3x3_kernel(const float* __restrict__ fm,
                                  float* __restrict__ out) {
  // [spatial(r,w)][channel] layout, channel-padded to 132 floats:
  // async-load side scatters (transpose), store side reads contiguous runs.
  __shared__ __align__(16) float lds[NSP * CPAD]; // 60192 B

  const int h   = blockIdx.x;   // output row 0..37
  const int ct  = blockIdx.y;   // channel tile 0..3
  const int b   = blockIdx.z;   // batch 0..15
  const int tid = threadIdx.x;
  const int c0  = ct * CT;

  const int hm = (h == 0)      ? 0      : h - 1;  // clamped row above
  const int hp = (h == HW - 1) ? HW - 1 : h + 1;  // clamped row below

  // Low 32 bits of the flat shared pointer == wave-relative LDS byte address
  // (ISA 10.2: LDS aperture maps addr[31:0] to LDS).
  const unsigned lds_base = (unsigned)(unsigned long long)(const void*)lds;

  const unsigned long long in_base =
      (unsigned long long)(const void*)fm +
      4ull * (unsigned long long)((b * CC + c0) * PP);
  const unsigned long long out_base =
      (unsigned long long)(void*)out +
      4ull * ((unsigned long long)(b * PP + h * HW) * OC + (unsigned long long)c0);

  // ---- Phase 1: async DMA global -> LDS, transposing to [spatial][channel] ----
  // Consecutive lanes walk consecutive w (contiguous global reads); each dword
  // lands at its own LDS address (the transpose). Uses ASYNCcnt, not VGPRs.
  for (int e = tid; e < NELEM; e += NTHREADS) {
    const int c    = e / NSP;          // 0..127
    const int rem  = e - c * NSP;      // 0..113  (= r*38 + w)
    const int r    = rem / HW;         // 0..2
    const int w    = rem - r * HW;     // 0..37
    const int hrow = (r == 0) ? hm : ((r == 1) ? h : hp);
    const unsigned long long gaddr =
        in_base + 4ull * (unsigned)(c * PP + hrow * HW + w);
    const unsigned laddr = lds_base + 4u * (unsigned)(rem * CPAD + c);
    asm volatile("global_load_async_to_lds_b32 %0, %1, off"
                 :: "v"(laddr), "v"(gaddr)
                 : "memory");
  }

  // Wait for this wave's async DMAs, then barrier so all LDS writes are visible.
#if __has_builtin(__builtin_amdgcn_s_wait_asynccnt)
  __builtin_amdgcn_s_wait_asynccnt(0);
#else
  asm volatile("s_wait_asynccnt 0" ::: "memory");
#endif
  __syncthreads();

  // ---- Phase 2: async DMA LDS -> global ----
  // For each (w', k) the 128 output channels are contiguous in LDS and in the
  // output; 32 lanes x b128 = perfectly coalesced 512 B bursts, no VGPR staging.
  for (int idx = tid; idx < NOUT; idx += NTHREADS) {
    const int c4 = idx & 31;        // 4-channel group
    const int t  = idx >> 5;
    const int k  = t % KK;          // neighbor 0..8 (row-major kh,kw)
    const int wp = t / KK;          // output col 0..37
    const int r  = k / 3;           // staged source row
    int ws = wp + (k - r * 3) - 1;  // clamped source col
    ws = ws < 0 ? 0 : (ws > HW - 1 ? HW - 1 : ws);
    const unsigned laddr =
        lds_base + 4u * (unsigned)((r * HW + ws) * CPAD) + 16u * (unsigned)c4;
    const unsigned long long gaddr =
        out_base + 4ull * (unsigned)(wp * OC + k * CC) + 16ull * (unsigned)c4;
    asm volatile("global_store_async_from_lds_b128 %0, %1, off"
                 :: "v"(gaddr), "v"(laddr)
                 : "memory");
  }
  // S_ENDPGM does an implicit wait-idle; outstanding async stores drain there.
}

extern "C" void kernel_launch(void* const* d_in, const int* in_sizes, int n_in,
                              void* d_out, int out_size, void* d_ws, size_t ws_size,
                              hipStream_t stream) {
  const float* fm = (const float*)d_in[0];
  float* out      = (float*)d_out;
  dim3 grid(HW, CC / CT, BB);   // 38 x 4 x 16 = 2432 blocks
  SpatialPool_gather3x3_kernel<<<grid, NTHREADS, 0, stream>>>(fm, out);
}